// RGCN_8435315769495
// MI455X (gfx1250) — compile-verified
//
#include <hip/hip_runtime.h>
#include <hip/hip_bf16.h>

typedef float v2f __attribute__((ext_vector_type(2)));
typedef float v8f __attribute__((ext_vector_type(8)));

#define D_DIM 128

// ---------------------------------------------------------------------------
// GEMM: support[n, o] = sum_k x[n,k] * W[k,o] + b[o]
// One wave computes one 16x16 output tile via 32 chained V_WMMA_F32_16X16X4_F32.
// Block = 256 threads = 8 waves -> covers all 128 output columns for 16 rows.
// grid.x = N/16 (N = 50000 = 16 * 3125 exactly).
//
// A-frag (16x4 f32, ISA 7.12.2): lanes 0-15 hold M=0..15 with K={k0,k0+1} in
// VGPR0/1; lanes 16-31 hold the same M rows with K={k0+2,k0+3}.
// B-frag (4x16 f32): mirror layout, lane&15 = N column, lane>>4 selects K pair.
// C/D (16x16 f32): VGPR g = row M=g (lanes 0-15) / M=g+8 (lanes 16-31),
// N = lane&15 (+ tile offset).
// ---------------------------------------------------------------------------
__global__ void rgcn_gemm_wmma(const float* __restrict__ x,
                               const float* __restrict__ W,   // [128,128] (k, o)
                               const float* __restrict__ b,   // [128]
                               float* __restrict__ support) { // [N,128]
    const int lane  = threadIdx.x & 31;
    const int wave  = threadIdx.x >> 5;     // 0..7 -> output-column tile
    const int row0  = blockIdx.x * 16;      // M tile base
    const int col0  = wave * 16;            // N tile base
    const int mrow  = lane & 15;            // M (for A) / N (for B/C/D) within tile
    const int khalf = (lane >> 4) * 2;      // 0 or 2 : which K pair this lane holds

    const float bias = b[col0 + mrow];
    v8f acc;
#pragma unroll
    for (int i = 0; i < 8; ++i) acc[i] = bias;   // C preloaded with bias (D = A*B + C)

    const float* __restrict__ xrow = x + (size_t)(row0 + mrow) * D_DIM;

#pragma unroll
    for (int k0 = 0; k0 < D_DIM; k0 += 4) {
        v2f a, bb;
        // A: two consecutive K values of this lane's row -> single b64 load
        a.x = xrow[k0 + khalf];
        a.y = xrow[k0 + khalf + 1];
        // B: W[k, col], strided by 128 floats between the two K values
        bb.x = W[(size_t)(k0 + khalf) * D_DIM + col0 + mrow];
        bb.y = W[(size_t)(k0 + khalf + 1) * D_DIM + col0 + mrow];
        // 8 args: (neg_a, A, neg_b, B, c_mod, C, reuse_a, reuse_b)
        acc = __builtin_amdgcn_wmma_f32_16x16x4_f32(
            false, a, false, bb, (short)0, acc, false, false);
    }

    const int hi = (lane >> 4) * 8;  // lanes 16-31 hold rows M=8..15
#pragma unroll
    for (int g = 0; g < 8; ++g) {
        support[(size_t)(row0 + g + hi) * D_DIM + (col0 + mrow)] = acc[g];
    }
}

// ---------------------------------------------------------------------------
// Edge scatter: one wave per edge (grid-stride over edges).
// Each lane handles 4 consecutive floats of the 128-float row:
//   gather support[src] (float4 per lane, b128), scale by edge_val,
//   4x native f32 atomic-add into out[dst].
// ---------------------------------------------------------------------------
__global__ void rgcn_edges(const float* __restrict__ support,
                           const int* __restrict__ esrc,
                           const int* __restrict__ edst,
                           const float* __restrict__ eval,
                           float* __restrict__ out,
                           int n_edges) {
    const int lane   = threadIdx.x & 31;
    const int waveId = (blockIdx.x * blockDim.x + threadIdx.x) >> 5;
    const int nWaves = (gridDim.x * blockDim.x) >> 5;

    for (int e = waveId; e < n_edges; e += nWaves) {
        const int   src = esrc[e];
        const int   dst = edst[e];
        const float v   = eval[e];

        // Prefetch next edge's source row into cache (global_prefetch_b8).
        const int en = e + nWaves;
        if (en < n_edges) {
            __builtin_prefetch((const void*)(support + (size_t)esrc[en] * D_DIM), 0, 1);
        }

        const float4* __restrict__ srow =
            (const float4*)(support + (size_t)src * D_DIM);
        float4 m = srow[lane];                       // 32 lanes x 4 floats = 128

        float* __restrict__ drow = out + (size_t)dst * D_DIM + lane * 4;
        unsafeAtomicAdd(drow + 0, m.x * v);
        unsafeAtomicAdd(drow + 1, m.y * v);
        unsafeAtomicAdd(drow + 2, m.z * v);
        unsafeAtomicAdd(drow + 3, m.w * v);
    }
}

__global__ void zero_f32(float* __restrict__ p, int n) {
    int i = blockIdx.x * blockDim.x + threadIdx.x;
    if (i < n) p[i] = 0.0f;
}

__global__ void tanh_f32(float* __restrict__ p, int n) {
    int i = blockIdx.x * blockDim.x + threadIdx.x;
    if (i < n) p[i] = tanhf(p[i]);
}

extern "C" void kernel_launch(void* const* d_in, const int* in_sizes, int n_in,
                              void* d_out, int out_size, void* d_ws, size_t ws_size,
                              hipStream_t stream) {
    // setup_inputs order: x, W, b, edge_val, edge_src, edge_dst
    const float* x    = (const float*)d_in[0];
    const float* W    = (const float*)d_in[1];
    const float* b    = (const float*)d_in[2];
    const float* eval = (const float*)d_in[3];
    const int*   esrc = (const int*)d_in[4];
    const int*   edst = (const int*)d_in[5];
    float*       out  = (float*)d_out;

    const int D = D_DIM;
    const int N = in_sizes[0] / D;          // 50000
    const int R = in_sizes[2] / D;          // 4
    const int E = in_sizes[3] / R;          // 800000

    float* support = (float*)d_ws;          // [N, 128] reused per relation (25.6 MB)

    const int nout = N * D;
    zero_f32<<<(nout + 255) / 256, 256, 0, stream>>>(out, nout);

    for (int r = 0; r < R; ++r) {
        rgcn_gemm_wmma<<<N / 16, 256, 0, stream>>>(
            x, W + (size_t)r * D * D, b + (size_t)r * D, support);
        rgcn_edges<<<4096, 256, 0, stream>>>(
            support, esrc + (size_t)r * E, edst + (size_t)r * E,
            eval + (size_t)r * E, out, E);
    }

    tanh_f32<<<(nout + 255) / 256, 256, 0, stream>>>(out, nout);
}